// Entropy_60129542144535
// MI455X (gfx1250) — compile-verified
//
#include <hip/hip_runtime.h>

// ---- CDNA5 vector types ----
typedef float v2f  __attribute__((ext_vector_type(2)));
typedef float v8f  __attribute__((ext_vector_type(8)));
typedef unsigned int u32x4 __attribute__((ext_vector_type(4)));
typedef int   i32x8 __attribute__((ext_vector_type(8)));
typedef int   i32x4 __attribute__((ext_vector_type(4)));

#define L_LEN 4096        // row length
#define KWIN  64          // window size
#define NWIN  4033        // L - k + 1 outputs per row
#define PAD   80          // LDS tail padding so last tile reads stay in-bounds
#define NTHREADS 256      // 8 waves (wave32)

// One block per (b,c) row. Phase 1: TDM DMA row -> LDS. Phase 2: e=exp(x),
// f=x*exp(x) into LDS. Phase 3: sliding-window-64 sums of e and f via
// banded-Toeplitz WMMA (V_WMMA_F32_16X16X4_F32), then out = T/S - log(S).
__global__ __launch_bounds__(NTHREADS) void entropy_kernel(
    const float* __restrict__ x, float* __restrict__ out)
{
    __shared__ float Xs[L_LEN];
    __shared__ float Es[L_LEN + PAD];
    __shared__ float Fs[L_LEN + PAD];

    const int row = blockIdx.x;                       // 0..255  (b*8 + c)
    const float* xrow = x + (size_t)row * L_LEN;
    float* orow = out + (size_t)row * NWIN;
    const int tid = threadIdx.x;

    // ---------------- Phase 1: Tensor Data Mover: DMA 4096 floats -> LDS ----
    if (tid < 32) {  // wave 0 only (wave-uniform branch)
        unsigned lds_off = (unsigned)(size_t)(&Xs[0]);          // LDS byte offset
        unsigned long long ga = (unsigned long long)(size_t)xrow;
        u32x4 g0;
        g0[0] = 1u;                                             // count=1, user desc
        g0[1] = lds_off;                                        // lds_addr
        g0[2] = (unsigned)(ga & 0xFFFFFFFFull);                 // global_addr lo
        g0[3] = (unsigned)((ga >> 32) & 0x1FFFFFFull) | (2u << 30); // addr hi | type=2
        i32x8 g1;
        g1[0] = 0x00020000;                                     // data_size=2 (4B)
        g1[1] = (int)(((unsigned)L_LEN & 0xFFFFu) << 16);       // tensor_dim0 lo16
        g1[2] = (int)(1u << 16);                                // tensor_dim1 = 1
        g1[3] = (int)(((unsigned)L_LEN & 0xFFFFu) << 16);       // tile_dim0 = 4096
        g1[4] = 1;                                              // tile_dim1 = 1
        g1[5] = L_LEN;                                          // dim0_stride lo32
        g1[6] = 0;
        g1[7] = 0;
        i32x4 gz4 = {0, 0, 0, 0};                               // groups 2/3 unused (<=2D)
        i32x8 gz8 = {0, 0, 0, 0, 0, 0, 0, 0};
        __builtin_amdgcn_tensor_load_to_lds(g0, g1, gz4, gz4, gz8, 0);
        __builtin_amdgcn_s_wait_tensorcnt(0);
    }
    __syncthreads();

    // ---------------- Phase 2: e = exp(x), f = x*exp(x) ---------------------
    for (int i = tid; i < L_LEN; i += NTHREADS) {
        float xv = Xs[i];
        float ev = __expf(xv);
        Es[i] = ev;
        Fs[i] = xv * ev;
    }
    for (int i = tid; i < PAD; i += NTHREADS) {   // zero tail (feeds only w>4032)
        Es[L_LEN + i] = 0.f;
        Fs[L_LEN + i] = 0.f;
    }
    __syncthreads();

    // ---------------- Phase 3: sliding-window-64 sums via WMMA --------------
    // Tile = 256 consecutive windows: M = offset 0..15, N = base step of 16.
    // S(w0+16n+m) = sum_{t=0..4} sum_{j'=0..15} A_t[m,j'] * e[w0+16n+16t+j']
    //   A_0 = upper-tri (j'>=m), A_1..3 = ones, A_4 = strict lower (j'<m).
    // Each 16x16x16 piece = 4x V_WMMA_F32_16X16X4_F32.
    const int lane = tid & 31;
    const int wave = tid >> 5;
    const int nIdx = lane & 15;       // B-matrix column N / A-matrix row M
    const int hi   = lane >> 4;       // 0 -> K rows {0,1}; 1 -> K rows {2,3}
    const int mA   = lane & 15;

    for (int tile = wave; tile < 16; tile += 8) {
        const int w0 = tile * 256;
        v8f accS = {0.f, 0.f, 0.f, 0.f, 0.f, 0.f, 0.f, 0.f};
        v8f accT = {0.f, 0.f, 0.f, 0.f, 0.f, 0.f, 0.f, 0.f};

        #pragma unroll
        for (int t = 0; t < 5; ++t) {
            const float* eb = &Es[w0 + 16 * nIdx + 16 * t];
            const float* fb = &Fs[w0 + 16 * nIdx + 16 * t];
            #pragma unroll
            for (int q = 0; q < 4; ++q) {
                const int jp = 4 * q + 2 * hi;     // this lane's local columns jp, jp+1
                v2f a, be, bf;
                if (t == 0) {
                    a[0] = (jp     >= mA) ? 1.f : 0.f;
                    a[1] = (jp + 1 >= mA) ? 1.f : 0.f;
                } else if (t == 4) {
                    a[0] = (jp     <  mA) ? 1.f : 0.f;
                    a[1] = (jp + 1 <  mA) ? 1.f : 0.f;
                } else {
                    a[0] = 1.f; a[1] = 1.f;
                }
                be[0] = eb[jp]; be[1] = eb[jp + 1];
                bf[0] = fb[jp]; bf[1] = fb[jp + 1];
                accS = __builtin_amdgcn_wmma_f32_16x16x4_f32(
                    false, a, false, be, (short)0, accS, false, false);
                accT = __builtin_amdgcn_wmma_f32_16x16x4_f32(
                    false, a, false, bf, (short)0, accT, false, false);
            }
        }

        // D layout: VGPR j -> M = j + 8*hi, N = lane&15  =>  8 consecutive w's.
        const int wbase = w0 + 16 * nIdx + 8 * hi;
        #pragma unroll
        for (int j = 0; j < 8; ++j) {
            const int w = wbase + j;
            if (w < NWIN) {
                float S = accS[j];
                float T = accT[j];
                orow[w] = T / S - __logf(S);
            }
        }
    }
}

extern "C" void kernel_launch(void* const* d_in, const int* in_sizes, int n_in,
                              void* d_out, int out_size, void* d_ws, size_t ws_size,
                              hipStream_t stream) {
    // d_in[0] = k (int scalar, ==64, baked into kernel), d_in[1] = x (32*8*4096 f32)
    const float* x = (const float*)d_in[1];
    float* out = (float*)d_out;   // 32*8*4033 f32
    entropy_kernel<<<dim3(256), dim3(NTHREADS), 0, stream>>>(x, out);
    (void)in_sizes; (void)n_in; (void)out_size; (void)d_ws; (void)ws_size;
}